// ARModel_1382979469861
// MI455X (gfx1250) — compile-verified
//
#include <hip/hip_runtime.h>
#include <hip/hip_bf16.h>

// AR(12) self-feeding recurrence, B=64, T=288, N=1024, P=12.
// Reformulated as a blocked linear scan: 16 steps at a time via per-node
// propagation matrices M_n (16x12) and bias vector d_n (16), executed with
// V_WMMA_F32_16X16X4_F32 (fp32 matrix pipe, matches reference precision).

typedef __attribute__((ext_vector_type(2))) float v2f;
typedef __attribute__((ext_vector_type(8))) float v8f;

#define B_ 64
#define T_ 288
#define N_ 1024
#define P_ 12
#define KBLK 16          // time steps per block
#define NTB (T_ / KBLK)  // 18 time blocks

// ---------------------------------------------------------------------------
// Phase 1: build propagation matrices.
//   M[n][k][p] = d y_{t+k} / d s_p   (s = last 12 outputs, p=0 oldest)
//   d[n][k]    = bias contribution accumulated over k steps
// One thread per (n, q): q<12 -> column p of M; q==12 -> bias column.
// ---------------------------------------------------------------------------
__global__ void ar_build_prop(const float* __restrict__ ar,
                              const float* __restrict__ bias,
                              float* __restrict__ Mmat,
                              float* __restrict__ dvec) {
  int tid = blockIdx.x * blockDim.x + threadIdx.x;
  if (tid >= N_ * 13) return;
  int n = tid / 13, q = tid % 13;

  float a[P_];
#pragma unroll
  for (int j = 0; j < P_; ++j) a[j] = ar[n * P_ + j];

  float ring[P_];  // column q of the last 12 rows of the unrolled map
#pragma unroll
  for (int i = 0; i < P_; ++i) ring[i] = (q < P_ && i == q) ? 1.0f : 0.0f;
  float c = (q == P_) ? bias[n] : 0.0f;

  for (int k = 0; k < KBLK; ++k) {
    float nv = c;
#pragma unroll
    for (int j = 0; j < P_; ++j) nv = fmaf(a[j], ring[j], nv);
    if (q < P_) Mmat[(n * KBLK + k) * P_ + q] = nv;
    else        dvec[n * KBLK + k] = nv;
#pragma unroll
    for (int i = 0; i < P_ - 1; ++i) ring[i] = ring[i + 1];
    ring[P_ - 1] = nv;
  }
}

// ---------------------------------------------------------------------------
// Phase 2: blocked scan with WMMA.
// Grid: (N/16, 2).  Block: 512 threads = 16 waves; wave w owns node n0+w and
// 32 batch columns (two 16-wide WMMA tiles). Per time block:
//   D(16t x 16b) = A(16t x 12s) * B(12s x 16b) + d   via 3x wmma f32 16x16x4
// LDS staging [k][bb][n_off] gives (a) coalesced stores with n contiguous
// across lanes and (b) the feedback state (rows 4..15 of D) for the next blk.
// ---------------------------------------------------------------------------
__global__ void __launch_bounds__(512)
ar_wmma_scan(const float* __restrict__ x, const float* __restrict__ Mmat,
             const float* __restrict__ dvec, float* __restrict__ out) {
  __shared__ float lds[KBLK * 32 * 16];  // 32 KB: [k][bb 0..31][n_off 0..15]

  const int tid = threadIdx.x;
  const int w   = tid >> 5;   // wave id = node offset (0..15)
  const int l   = tid & 31;
  const int col = l & 15;     // A row m / B-C-D column
  const int hal = l >> 4;     // lane half per ISA fragment layouts
  const int n0  = blockIdx.x * 16;
  const int n   = n0 + w;
  const long b_base = (long)blockIdx.y * 32;

  // A fragments (ISA 16x4 f32 A: VGPRv holds K = 4c + 2*hal + v, row = l%16)
  v2f afr[3];
#pragma unroll
  for (int c = 0; c < 3; ++c) {
    afr[c].x = Mmat[(n * KBLK + col) * P_ + (4 * c + 2 * hal + 0)];
    afr[c].y = Mmat[(n * KBLK + col) * P_ + (4 * c + 2 * hal + 1)];
  }
  // C init = bias propagation (C layout: VGPR r <-> row r + 8*hal)
  float dini[8];
#pragma unroll
  for (int r = 0; r < 8; ++r) dini[r] = dvec[n * KBLK + r + 8 * hal];

  // Initial state from tail of x: s_p = x[b][T-12+p][n]
  v2f bfr[2][3];
#pragma unroll
  for (int j = 0; j < 2; ++j) {
    long b = b_base + j * 16 + col;
#pragma unroll
    for (int c = 0; c < 3; ++c) {
      int k0 = 4 * c + 2 * hal;
      bfr[j][c].x = x[(b * T_ + (T_ - P_ + k0 + 0)) * N_ + n];
      bfr[j][c].y = x[(b * T_ + (T_ - P_ + k0 + 1)) * N_ + n];
    }
  }

  for (int tb = 0; tb < NTB; ++tb) {
    v8f acc[2];
#pragma unroll
    for (int j = 0; j < 2; ++j) {
#pragma unroll
      for (int r = 0; r < 8; ++r) acc[j][r] = dini[r];
#pragma unroll
      for (int c = 0; c < 3; ++c)
        acc[j] = __builtin_amdgcn_wmma_f32_16x16x4_f32(
            false, afr[c], false, bfr[j][c], (short)0, acc[j], false, false);
    }

    __syncthreads();  // previous block's LDS readers are done
#pragma unroll
    for (int j = 0; j < 2; ++j) {
      int bb = j * 16 + col;
#pragma unroll
      for (int r = 0; r < 8; ++r) {
        int k = r + 8 * hal;
        lds[(k * 32 + bb) * 16 + w] = acc[j][r];
      }
    }
    __syncthreads();

    // Coalesced store: lanes 0..15 cover 16 consecutive n (64B segments)
    {
      int n_off = tid & 15;
      int bb    = tid >> 4;                 // 0..31
      long b    = b_base + bb;
      long base = (b * T_ + (long)tb * KBLK) * N_ + n0 + n_off;
#pragma unroll
      for (int k = 0; k < KBLK; ++k)
        out[base + (long)k * N_] = lds[(k * 32 + bb) * 16 + n_off];
    }

    // Feedback: next state s'[p] = y_{t+4+p} = D rows 4..15 (from LDS)
#pragma unroll
    for (int j = 0; j < 2; ++j) {
      int bb = j * 16 + col;
#pragma unroll
      for (int c = 0; c < 3; ++c) {
        int k0 = 4 + 4 * c + 2 * hal;
        bfr[j][c].x = lds[((k0 + 0) * 32 + bb) * 16 + w];
        bfr[j][c].y = lds[((k0 + 1) * 32 + bb) * 16 + w];
      }
    }
  }
}

// ---------------------------------------------------------------------------
extern "C" void kernel_launch(void* const* d_in, const int* in_sizes, int n_in,
                              void* d_out, int out_size, void* d_ws, size_t ws_size,
                              hipStream_t stream) {
  const float* x    = (const float*)d_in[0];  // (B, T, N, 1) f32
  const float* ar   = (const float*)d_in[1];  // (N, 12) f32
  const float* bias = (const float*)d_in[2];  // (N,) f32
  float* out = (float*)d_out;                 // (B, T, N, 1) f32

  // Workspace: M (N*16*12 f32) + d (N*16 f32) = ~832 KB, rebuilt every call.
  float* Mmat = (float*)d_ws;
  float* dvec = Mmat + (size_t)N_ * KBLK * P_;

  ar_build_prop<<<(N_ * 13 + 255) / 256, 256, 0, stream>>>(ar, bias, Mmat, dvec);

  dim3 grid(N_ / 16, 2);  // 64 node-chunks x 2 batch halves
  ar_wmma_scan<<<grid, 512, 0, stream>>>(x, Mmat, dvec, out);
}